// GNN_23407571763485
// MI455X (gfx1250) — compile-verified
//
#include <hip/hip_runtime.h>
#include <hip/hip_bf16.h>

typedef __attribute__((ext_vector_type(2))) float v2f;
typedef __attribute__((ext_vector_type(8))) float v8f;

#define N_NODES 100000
#define D_FEAT 256

// ---------------- degree / normalization ----------------

__global__ void gnn_init_deg(float* deg, int n) {
    int i = blockIdx.x * blockDim.x + threadIdx.x;
    if (i < n) deg[i] = 1.0f;  // self-loop
}

__global__ void gnn_count_deg(const int* __restrict__ dst, float* deg, int e) {
    int i = blockIdx.x * blockDim.x + threadIdx.x;
    if (i < e) atomicAdd(&deg[dst[i]], 1.0f);
}

__global__ void gnn_dinv(const float* __restrict__ deg, float* dinv, int n) {
    int i = blockIdx.x * blockDim.x + threadIdx.x;
    if (i < n) dinv[i] = rsqrtf(deg[i]);  // deg >= 1 always
}

// ---------------- dense transform: WMMA f32 16x16x4 ----------------
// A: M x K row-major. W: K x NCOL row-major. Out: M x NCOL row-major.
// One wave computes a 16x16 output tile; NCOL/16 waves per block cover one
// 16-row block. M must be a multiple of 16 (100000 = 6250 * 16).
template <int K, int NCOL, bool RELU_A>
__global__ void gnn_gemm_wmma(const float* __restrict__ A,
                              const float* __restrict__ W,
                              float* __restrict__ Out, int M) {
    const int wave = threadIdx.x >> 5;
    const int lane = threadIdx.x & 31;
    const int rowBase = blockIdx.x * 16;
    const int colBase = wave * 16;
    if (rowBase >= M) return;

    const int r = lane & 15;            // A: row-in-tile; B/D: col-in-tile
    const int khalf = (lane >> 4) * 2;  // lane half selects K pair {0,1} / {2,3}

    v8f acc = {};
    const float* arow = A + (size_t)(rowBase + r) * K + khalf;
#pragma unroll 4
    for (int k0 = 0; k0 < K; k0 += 4) {
        v2f a, b;
        float a0 = arow[k0];
        float a1 = arow[k0 + 1];
        if (RELU_A) { a0 = fmaxf(a0, 0.0f); a1 = fmaxf(a1, 0.0f); }
        a.x = a0; a.y = a1;
        const float* bp = W + (size_t)(k0 + khalf) * NCOL + colBase + r;
        b.x = bp[0];
        b.y = bp[NCOL];
        // D = A*B + C   (exact f32 math)
        acc = __builtin_amdgcn_wmma_f32_16x16x4_f32(
            /*neg_a=*/false, a, /*neg_b=*/false, b,
            /*c_mod=*/(short)0, acc, /*reuse_a=*/false, /*reuse_b=*/false);
    }

    // C/D layout: VGPR g -> M = g (+8 for upper lane half), N = lane&15
    const int mOff = (lane >> 4) * 8;
    const int col = colBase + r;
#pragma unroll
    for (int g = 0; g < 8; ++g) {
        Out[(size_t)(rowBase + mOff + g) * NCOL + col] = acc[g];
    }
}

// ---------------- bias init + edge scatter ----------------

template <int F>
__global__ void gnn_init_bias(float* out, const float* __restrict__ b, int n) {
    long long i = (long long)blockIdx.x * blockDim.x + threadIdx.x;
    long long total = (long long)n * F;
    if (i < total) out[i] = b[(int)(i % F)];
}

// Messages 0..E-1 are edges, E..E+N-1 are self-loops.
// Thread layout: consecutive threads = consecutive features of one message,
// so the gather read of hw[src] is a contiguous F*4-byte burst.
template <int F>
__global__ void gnn_scatter(const float* __restrict__ hw,
                            const int* __restrict__ src,
                            const int* __restrict__ dst,
                            const float* __restrict__ dinv,
                            float* out, int E, int N) {
    long long idx = (long long)blockIdx.x * blockDim.x + threadIdx.x;
    long long total = (long long)(E + N) * F;
    if (idx >= total) return;
    int e = (int)(idx / F);
    int f = (int)(idx % F);
    int s, d;
    if (e < E) { s = src[e]; d = dst[e]; }
    else       { s = d = e - E; }
    float nrm = dinv[s] * dinv[d];
    atomicAdd(&out[(size_t)d * F + f], hw[(size_t)s * F + f] * nrm);
}

// ---------------- final MLP: relu(h@wf1+bf1)@wf2+bf2 ----------------

__global__ void gnn_mlp(const float* __restrict__ agg,  // N x 32 (pre-relu)
                        const float* __restrict__ wf1,  // 32 x 16
                        const float* __restrict__ bf1,  // 16
                        const float* __restrict__ wf2,  // 16 x 1
                        const float* __restrict__ bf2,  // 1
                        float* __restrict__ out, int n) {
    int i = blockIdx.x * blockDim.x + threadIdx.x;
    if (i >= n) return;
    float h[32];
#pragma unroll
    for (int j = 0; j < 32; ++j) h[j] = fmaxf(agg[(size_t)i * 32 + j], 0.0f);
    float acc = bf2[0];
#pragma unroll
    for (int k = 0; k < 16; ++k) {
        float t = bf1[k];
#pragma unroll
        for (int j = 0; j < 32; ++j) t += h[j] * wf1[j * 16 + k];
        acc += fmaxf(t, 0.0f) * wf2[k];
    }
    out[i] = acc;
}

// ---------------- launch ----------------

extern "C" void kernel_launch(void* const* d_in, const int* in_sizes, int n_in,
                              void* d_out, int out_size, void* d_ws, size_t ws_size,
                              hipStream_t stream) {
    const float* x   = (const float*)d_in[0];
    const int*   ei  = (const int*)  d_in[1];
    const float* w1  = (const float*)d_in[2];
    const float* b1  = (const float*)d_in[3];
    const float* w2  = (const float*)d_in[4];
    const float* b2  = (const float*)d_in[5];
    const float* wf1 = (const float*)d_in[6];
    const float* bf1 = (const float*)d_in[7];
    const float* wf2 = (const float*)d_in[8];
    const float* bf2 = (const float*)d_in[9];
    float* out = (float*)d_out;

    const int N = N_NODES;
    const int E = in_sizes[1] / 2;  // 1,600,000
    const int* src = ei;
    const int* dst = ei + E;

    char* ws = (char*)d_ws;
    float* deg  = (float*)ws; ws += (size_t)N * sizeof(float);
    float* dinv = (float*)ws; ws += (size_t)N * sizeof(float);
    float* hw1  = (float*)ws; ws += (size_t)N * 64 * sizeof(float);
    float* agg1 = (float*)ws; ws += (size_t)N * 64 * sizeof(float);
    float* hw2  = (float*)ws; ws += (size_t)N * 32 * sizeof(float);
    float* agg2 = (float*)ws; ws += (size_t)N * 32 * sizeof(float);

    // normalization
    gnn_init_deg<<<(N + 255) / 256, 256, 0, stream>>>(deg, N);
    gnn_count_deg<<<(E + 255) / 256, 256, 0, stream>>>(dst, deg, E);
    gnn_dinv<<<(N + 255) / 256, 256, 0, stream>>>(deg, dinv, N);

    // layer 1: hw1 = x @ w1 ; agg1 = scatter(norm * hw1[src]) + b1
    gnn_gemm_wmma<D_FEAT, 64, false><<<N / 16, 128, 0, stream>>>(x, w1, hw1, N);
    gnn_init_bias<64><<<(int)(((long long)N * 64 + 255) / 256), 256, 0, stream>>>(agg1, b1, N);
    {
        long long total = (long long)(E + N) * 64;
        gnn_scatter<64><<<(int)((total + 255) / 256), 256, 0, stream>>>(
            hw1, src, dst, dinv, agg1, E, N);
    }

    // layer 2: hw2 = relu(agg1) @ w2 ; agg2 = scatter(norm * hw2[src]) + b2
    gnn_gemm_wmma<64, 32, true><<<N / 16, 64, 0, stream>>>(agg1, w2, hw2, N);
    gnn_init_bias<32><<<(int)(((long long)N * 32 + 255) / 256), 256, 0, stream>>>(agg2, b2, N);
    {
        long long total = (long long)(E + N) * 32;
        gnn_scatter<32><<<(int)((total + 255) / 256), 256, 0, stream>>>(
            hw2, src, dst, dinv, agg2, E, N);
    }

    // head
    gnn_mlp<<<(N + 255) / 256, 256, 0, stream>>>(agg2, wf1, bf1, wf2, bf2, out, N);
}